// DPPAttention_541165879779
// MI455X (gfx1250) — compile-verified
//
#include <hip/hip_runtime.h>
#include <hip/hip_bf16.h>
#include <math.h>

// ---------------------------------------------------------------------------
// DPP attention pipeline for MI455X (gfx1250, wave32, WMMA f32 16x16x4).
// All GEMMs run through v_wmma_f32_16x16x4_f32 (memory-bound workload ->
// fp32 matrix path keeps reference precision at no cost).
// ---------------------------------------------------------------------------

typedef __attribute__((ext_vector_type(2))) float v2f;
typedef __attribute__((ext_vector_type(8))) float v8f;

#define B_      1024
#define S_      100
#define H_      512
#define SUB_EPS 1e-5f
#define LN_EPS  1e-12f

__device__ __forceinline__ v8f wmma4(v2f a, v2f b, v8f c) {
  // 8 args: (neg_a, A, neg_b, B, c_mod, C, reuse_a, reuse_b)
  return __builtin_amdgcn_wmma_f32_16x16x4_f32(false, a, false, b, (short)0, c,
                                               false, false);
}

// ---------------------------------------------------------------------------
// Kernel 1: qsq = (X @ Wq + bq)^2        [102400 x 512] x [512 x 512]
// One wave per 16(M) x 64(N) strip; M=102400 and N=512 divide exactly.
// ---------------------------------------------------------------------------
__global__ __launch_bounds__(256) void k_proj_sq(
    const float* __restrict__ X, const float* __restrict__ Wq,
    const float* __restrict__ bq, float* __restrict__ qsq) {
  const int lane = threadIdx.x & 31;
  const int wave = threadIdx.x >> 5;
  const int task = blockIdx.x * 8 + wave;   // 51200 wave tasks
  const int mt = task >> 3;                 // 6400 M tiles
  const int ns = task & 7;                  // 8 N strips
  const int m0 = mt * 16;
  const int n0 = ns * 64;
  const int half = lane >> 4;
  const int lm = lane & 15;

  const float* arow = X + (size_t)(m0 + lm) * H_ + 2 * half;
  v8f acc0 = {}, acc1 = {}, acc2 = {}, acc3 = {};
  for (int k = 0; k < H_; k += 4) {
    __builtin_prefetch(arow + k + 64, 0, 0);
    v2f a = *(const v2f*)(arow + k);
    const float* bp0 = Wq + (size_t)(k + 2 * half) * H_ + n0 + lm;
    const float* bp1 = bp0 + H_;
    v2f b;
    b.x = bp0[0];  b.y = bp1[0];  acc0 = wmma4(a, b, acc0);
    b.x = bp0[16]; b.y = bp1[16]; acc1 = wmma4(a, b, acc1);
    b.x = bp0[32]; b.y = bp1[32]; acc2 = wmma4(a, b, acc2);
    b.x = bp0[48]; b.y = bp1[48]; acc3 = wmma4(a, b, acc3);
  }
  const int c0 = n0 + lm;
  const float q0 = bq[c0], q1 = bq[c0 + 16], q2 = bq[c0 + 32], q3 = bq[c0 + 48];
#pragma unroll
  for (int i = 0; i < 8; ++i) {
    const int row = m0 + i + 8 * half;
    float* o = qsq + (size_t)row * H_ + c0;
    float r;
    r = acc0[i] + q0; o[0]  = r * r;
    r = acc1[i] + q1; o[16] = r * r;
    r = acc2[i] + q2; o[32] = r * r;
    r = acc3[i] + q3; o[48] = r * r;
  }
}

// ---------------------------------------------------------------------------
// Kernel 2: G[b] = qsq[b] @ qsq[b]^T     [100 x 512] x [512 x 100]
// One wave per (batch, 16-row tile) computing a 16 x 112 strip (7 j-tiles).
// B = A^T, so both fragments are contiguous b64 loads along H.
// ---------------------------------------------------------------------------
__global__ __launch_bounds__(256) void k_gram(
    const float* __restrict__ qsq, float* __restrict__ G) {
  const int lane = threadIdx.x & 31;
  const int wave = threadIdx.x >> 5;
  const int task = blockIdx.x * 8 + wave;   // 1024*7 = 7168
  const int b = task / 7;
  const int it = task - b * 7;
  const int i0 = it * 16;
  const int half = lane >> 4, lm = lane & 15;
  const float* Q = qsq + (size_t)b * S_ * H_;

  const int ar = i0 + lm;
  const bool av = ar < S_;
  const float* ap = Q + (size_t)(av ? ar : 0) * H_ + 2 * half;

  bool bv[7];
  const float* bp[7];
#pragma unroll
  for (int j = 0; j < 7; ++j) {
    const int col = j * 16 + lm;
    bv[j] = col < S_;
    bp[j] = Q + (size_t)(bv[j] ? col : 0) * H_ + 2 * half;
  }

  v8f acc[7] = {};
  v2f zero = {};
  for (int k = 0; k < H_; k += 4) {
    v2f a = av ? *(const v2f*)(ap + k) : zero;
#pragma unroll
    for (int j = 0; j < 7; ++j) {
      v2f bb = bv[j] ? *(const v2f*)(bp[j] + k) : zero;
      acc[j] = wmma4(a, bb, acc[j]);
    }
  }
#pragma unroll
  for (int i = 0; i < 8; ++i) {
    const int row = i0 + i + 8 * half;
    if (row < S_) {
      float* g = G + ((size_t)b * S_ + row) * S_;
#pragma unroll
      for (int j = 0; j < 7; ++j) {
        const int col = j * 16 + lm;
        if (col < S_) g[col] = acc[j][i];
      }
    }
  }
}

// ---------------------------------------------------------------------------
// Kernel 3a: per batch -> gd (diag of G) and denom (strict-upper det sum)
// ---------------------------------------------------------------------------
__global__ __launch_bounds__(256) void k_denom(
    const float* __restrict__ G, float* __restrict__ gd,
    float* __restrict__ denom) {
  const int b = blockIdx.x;
  const float* Gb = G + (size_t)b * (S_ * S_);
  __shared__ float sgd[S_];
  for (int i = threadIdx.x; i < S_; i += 256) sgd[i] = Gb[i * S_ + i];
  __syncthreads();

  float sum = 0.f;
  for (int idx = threadIdx.x; idx < S_ * S_; idx += 256) {
    const int i = idx / S_;
    const int j = idx - i * S_;
    if (j > i) {
      const float gij = Gb[idx];
      sum += (sgd[i] + SUB_EPS) * (sgd[j] + SUB_EPS) - gij * gij;
    }
  }
#pragma unroll
  for (int off = 16; off; off >>= 1) sum += __shfl_xor(sum, off, 32);
  __shared__ float red[8];
  if ((threadIdx.x & 31) == 0) red[threadIdx.x >> 5] = sum;
  __syncthreads();
  if (threadIdx.x < 8) {
    float v = red[threadIdx.x];
    v += __shfl_xor(v, 4, 32);
    v += __shfl_xor(v, 2, 32);
    v += __shfl_xor(v, 1, 32);
    if (threadIdx.x == 0) denom[b] = v;
  }
  for (int i = threadIdx.x; i < S_; i += 256) gd[(size_t)b * S_ + i] = sgd[i];
}

// ---------------------------------------------------------------------------
// Kernel 3b: row softmax of -(det/denom + diag)/8 + mask, in place on G.
// One wave per row; wave32 shfl reductions.
// ---------------------------------------------------------------------------
__global__ __launch_bounds__(256) void k_softmax(
    float* __restrict__ G, const float* __restrict__ gd,
    const float* __restrict__ denom, const float* __restrict__ mask) {
  const int lane = threadIdx.x & 31;
  const int wave = threadIdx.x >> 5;
  const int row = blockIdx.x * 8 + wave;    // B*S rows
  const int b = row / S_;
  const int i = row - b * S_;
  float* Gr = G + (size_t)row * S_;
  const float* gdb = gd + (size_t)b * S_;
  const float* mb = mask + (size_t)b * S_;
  const float gi = gdb[i];
  const float giE = gi + SUB_EPS;
  const float inv_den = 1.f / denom[b];

  float v[4];
  float mx = -1e30f;
#pragma unroll
  for (int t = 0; t < 4; ++t) {
    const int j = lane + t * 32;
    float sc = -1e30f;
    if (j < S_) {
      const float gij = Gr[j];
      const float det = giE * (gdb[j] + SUB_EPS) - gij * gij;
      sc = -(det * inv_den + (j == i ? gi : 0.f)) * 0.125f + mb[j];
    }
    v[t] = sc;
    mx = fmaxf(mx, sc);
  }
#pragma unroll
  for (int off = 16; off; off >>= 1) mx = fmaxf(mx, __shfl_xor(mx, off, 32));
  float sum = 0.f;
#pragma unroll
  for (int t = 0; t < 4; ++t) { v[t] = __expf(v[t] - mx); sum += v[t]; }
#pragma unroll
  for (int off = 16; off; off >>= 1) sum += __shfl_xor(sum, off, 32);
  const float inv = 1.f / sum;
#pragma unroll
  for (int t = 0; t < 4; ++t) {
    const int j = lane + t * 32;
    if (j < S_) Gr[j] = v[t] * inv;
  }
}

// ---------------------------------------------------------------------------
// Kernel 4: ctx[b] = probs[b] @ X[b]     [100 x 100] x [100 x 512]
// K=100 divides by 4 exactly; only M edge needs guards.
// ---------------------------------------------------------------------------
__global__ __launch_bounds__(256) void k_ctx(
    const float* __restrict__ P, const float* __restrict__ X,
    float* __restrict__ ctx) {
  const int lane = threadIdx.x & 31;
  const int wave = threadIdx.x >> 5;
  const int task = blockIdx.x * 8 + wave;   // 1024*7*8 = 57344
  const int ns = task & 7;
  const int t2 = task >> 3;
  const int b = t2 / 7;
  const int it = t2 - b * 7;
  const int i0 = it * 16, n0 = ns * 64;
  const int half = lane >> 4, lm = lane & 15;

  const float* Pb = P + (size_t)b * (S_ * S_);
  const float* Xb = X + (size_t)b * S_ * H_;
  const int ar = i0 + lm;
  const bool av = ar < S_;
  const float* ap = Pb + (size_t)(av ? ar : 0) * S_ + 2 * half;

  v8f acc0 = {}, acc1 = {}, acc2 = {}, acc3 = {};
  v2f zero = {};
  for (int k = 0; k < S_; k += 4) {
    v2f a = av ? *(const v2f*)(ap + k) : zero;
    const float* bp0 = Xb + (size_t)(k + 2 * half) * H_ + n0 + lm;
    const float* bp1 = bp0 + H_;
    v2f bb;
    bb.x = bp0[0];  bb.y = bp1[0];  acc0 = wmma4(a, bb, acc0);
    bb.x = bp0[16]; bb.y = bp1[16]; acc1 = wmma4(a, bb, acc1);
    bb.x = bp0[32]; bb.y = bp1[32]; acc2 = wmma4(a, bb, acc2);
    bb.x = bp0[48]; bb.y = bp1[48]; acc3 = wmma4(a, bb, acc3);
  }
#pragma unroll
  for (int i = 0; i < 8; ++i) {
    const int row = i0 + i + 8 * half;
    if (row < S_) {
      float* o = ctx + ((size_t)b * S_ + row) * H_ + n0 + lm;
      o[0]  = acc0[i];
      o[16] = acc1[i];
      o[32] = acc2[i];
      o[48] = acc3[i];
    }
  }
}

// ---------------------------------------------------------------------------
// Kernel 5: out = ctx @ Wd + bd + X  (staged into d_out, LN follows)
// ---------------------------------------------------------------------------
__global__ __launch_bounds__(256) void k_out_proj(
    const float* __restrict__ ctx, const float* __restrict__ Wd,
    const float* __restrict__ bd, const float* __restrict__ X,
    float* __restrict__ out) {
  const int lane = threadIdx.x & 31;
  const int wave = threadIdx.x >> 5;
  const int task = blockIdx.x * 8 + wave;
  const int mt = task >> 3;
  const int ns = task & 7;
  const int m0 = mt * 16;
  const int n0 = ns * 64;
  const int half = lane >> 4;
  const int lm = lane & 15;

  const float* arow = ctx + (size_t)(m0 + lm) * H_ + 2 * half;
  v8f acc0 = {}, acc1 = {}, acc2 = {}, acc3 = {};
  for (int k = 0; k < H_; k += 4) {
    __builtin_prefetch(arow + k + 64, 0, 0);
    v2f a = *(const v2f*)(arow + k);
    const float* bp0 = Wd + (size_t)(k + 2 * half) * H_ + n0 + lm;
    const float* bp1 = bp0 + H_;
    v2f b;
    b.x = bp0[0];  b.y = bp1[0];  acc0 = wmma4(a, b, acc0);
    b.x = bp0[16]; b.y = bp1[16]; acc1 = wmma4(a, b, acc1);
    b.x = bp0[32]; b.y = bp1[32]; acc2 = wmma4(a, b, acc2);
    b.x = bp0[48]; b.y = bp1[48]; acc3 = wmma4(a, b, acc3);
  }
  const int c0 = n0 + lm;
  const float d0 = bd[c0], d1 = bd[c0 + 16], d2 = bd[c0 + 32], d3 = bd[c0 + 48];
#pragma unroll
  for (int i = 0; i < 8; ++i) {
    const int row = m0 + i + 8 * half;
    const float* xr = X + (size_t)row * H_ + c0;
    float* o = out + (size_t)row * H_ + c0;
    o[0]  = acc0[i] + d0 + xr[0];
    o[16] = acc1[i] + d1 + xr[16];
    o[32] = acc2[i] + d2 + xr[32];
    o[48] = acc3[i] + d3 + xr[48];
  }
}

// ---------------------------------------------------------------------------
// Kernel 6: LayerNorm in place on d_out; one wave per 512-wide row.
// ---------------------------------------------------------------------------
__global__ __launch_bounds__(256) void k_ln(
    float* __restrict__ out, const float* __restrict__ w,
    const float* __restrict__ bias) {
  const int lane = threadIdx.x & 31;
  const int wave = threadIdx.x >> 5;
  const int row = blockIdx.x * 8 + wave;
  float* r = out + (size_t)row * H_;
  float x[16];
  float s = 0.f;
#pragma unroll
  for (int t = 0; t < 16; ++t) { x[t] = r[lane + t * 32]; s += x[t]; }
#pragma unroll
  for (int off = 16; off; off >>= 1) s += __shfl_xor(s, off, 32);
  const float u = s * (1.f / 512.f);
  float var = 0.f;
#pragma unroll
  for (int t = 0; t < 16; ++t) { const float d = x[t] - u; var += d * d; }
#pragma unroll
  for (int off = 16; off; off >>= 1) var += __shfl_xor(var, off, 32);
  const float rinv = rsqrtf(var * (1.f / 512.f) + LN_EPS);
#pragma unroll
  for (int t = 0; t < 16; ++t) {
    const int c = lane + t * 32;
    r[c] = w[c] * (x[t] - u) * rinv + bias[c];
  }
}

// ---------------------------------------------------------------------------
extern "C" void kernel_launch(void* const* d_in, const int* in_sizes, int n_in,
                              void* d_out, int out_size, void* d_ws,
                              size_t ws_size, hipStream_t stream) {
  (void)in_sizes; (void)n_in; (void)out_size; (void)ws_size;
  const float* X    = (const float*)d_in[0];
  const float* mask = (const float*)d_in[2];   // [b,1,s]
  const float* Wq   = (const float*)d_in[5];
  const float* bq   = (const float*)d_in[6];
  const float* Wd   = (const float*)d_in[7];
  const float* bd   = (const float*)d_in[8];
  const float* lnw  = (const float*)d_in[9];
  const float* lnb  = (const float*)d_in[10];
  float* out = (float*)d_out;

  // Workspace layout (~251 MB): qsq/ctx share, G/probs share (in place).
  float* qsq = (float*)d_ws;                      // B*S*H floats
  float* G   = qsq + (size_t)B_ * S_ * H_;        // B*S*S floats
  float* gd  = G + (size_t)B_ * S_ * S_;          // B*S floats
  float* den = gd + (size_t)B_ * S_;              // B floats

  k_proj_sq<<<6400, 256, 0, stream>>>(X, Wq, bq, qsq);
  k_gram<<<896, 256, 0, stream>>>(qsq, G);
  k_denom<<<B_, 256, 0, stream>>>(G, gd, den);
  k_softmax<<<12800, 256, 0, stream>>>(G, gd, den, mask);
  k_ctx<<<7168, 256, 0, stream>>>(G, X, qsq);          // ctx overwrites qsq
  k_out_proj<<<6400, 256, 0, stream>>>(qsq, Wd, bd, X, out);
  k_ln<<<12800, 256, 0, stream>>>(out, lnw, lnb);
}